// NetworkFlowGCN_40913858462144
// MI455X (gfx1250) — compile-verified
//
#include <hip/hip_runtime.h>
#include <math.h>

typedef float v2f __attribute__((ext_vector_type(2)));
typedef float v8f __attribute__((ext_vector_type(8)));

#define D_IN 128
#define EPS 1e-5f
#define NEG_SLOPE 0.2f

// ---------- helpers ----------
__device__ __forceinline__ unsigned ordenc(float f) {
    unsigned u = __float_as_uint(f);
    return (u & 0x80000000u) ? ~u : (u | 0x80000000u);
}
__device__ __forceinline__ float orddec(unsigned u) {
    return __uint_as_float((u & 0x80000000u) ? (u & 0x7FFFFFFFu) : ~u);
}

// ---------- degree / normalization ----------
__global__ void k_init_deg(unsigned* deg, int N) {
    int i = blockIdx.x * blockDim.x + threadIdx.x;
    if (i < N) deg[i] = 1u;                 // self loop
}
__global__ void k_count_deg(const int* __restrict__ dst, unsigned* deg, int E) {
    int i = blockIdx.x * blockDim.x + threadIdx.x;
    if (i < E) atomicAdd(&deg[dst[i]], 1u);
}
__global__ void k_deg_to_dis(float* dis, int N) {
    int i = blockIdx.x * blockDim.x + threadIdx.x;
    if (i < N) {
        unsigned d = ((const unsigned*)dis)[i];
        dis[i] = rsqrtf((float)d);
    }
}

__global__ void k_zero(float* p, long long n) {
    long long i = blockIdx.x * (long long)blockDim.x + threadIdx.x;
    if (i < n) p[i] = 0.0f;
}

// ---------- WMMA fp32 GEMM: C[M,F] = A[M,K] @ B[K,F], row major, M%16==0, K%4==0, F%16==0
// B (<= 128x64 = 32KB) is staged once per block into LDS and shared by all 8 waves.
__global__ void k_wmma_gemm_f32(const float* __restrict__ A, const float* __restrict__ B,
                                float* __restrict__ C, int Mtiles, int K, int F) {
    __shared__ float Bs[128 * 64];

    // cooperative float4 stage of B into LDS (K*F is a multiple of 4)
    int quads = (K * F) >> 2;
    for (int i = threadIdx.x; i < quads; i += blockDim.x)
        ((float4*)Bs)[i] = ((const float4*)B)[i];
    __syncthreads();

    int wave = blockIdx.x * (blockDim.x >> 5) + (threadIdx.x >> 5);
    int ftiles = F >> 4;
    int mt = wave / ftiles;
    int nt = wave % ftiles;
    if (mt >= Mtiles) return;               // wave-uniform exit, EXEC stays all-ones

    int lane = threadIdx.x & 31;
    int half = lane >> 4;                   // 0: K={k0,k0+1}  1: K={k0+2,k0+3}
    int ls   = lane & 15;

    const float* arow = A + (long long)(mt * 16 + ls) * K;
    const float* bcol = Bs + nt * 16 + ls;

    v8f acc = (v8f)(0.0f);
    for (int k0 = 0; k0 < K; k0 += 4) {
        int k = k0 + half * 2;
        v2f a, b;
        a.x = arow[k];
        a.y = arow[k + 1];
        b.x = bcol[k * F];
        b.y = bcol[(k + 1) * F];
        acc = __builtin_amdgcn_wmma_f32_16x16x4_f32(false, a, false, b,
                                                    (short)0, acc, false, false);
    }
    // C/D layout: VGPR r -> row (half*8 + r), col = ls
    float* crow = C + (long long)(mt * 16 + half * 8) * F + nt * 16 + ls;
#pragma unroll
    for (int r = 0; r < 8; ++r) crow[(long long)r * F] = acc[r];
}

// ---------- GCN message passing: out[dst] += h[src] * dis[src]*dis[dst], + self loops
__global__ void k_gcn_scatter(const float* __restrict__ h, const int* __restrict__ src,
                              const int* __restrict__ dst, const float* __restrict__ dis,
                              float* __restrict__ out, int F4, int E, int N) {
    long long t = blockIdx.x * (long long)blockDim.x + threadIdx.x;
    long long total = (long long)(E + N) * F4;
    if (t >= total) return;
    int e = (int)(t / F4);
    int c = (int)(t % F4);
    int s, d;
    if (e < E) { s = src[e]; d = dst[e]; } else { s = d = e - E; }
    float nrm = dis[s] * dis[d];
    int F = F4 * 4;
    float4 v = *(const float4*)(h + (long long)s * F + c * 4);
    float* o = out + (long long)d * F + c * 4;
    atomicAdd(o + 0, v.x * nrm);
    atomicAdd(o + 1, v.y * nrm);
    atomicAdd(o + 2, v.z * nrm);
    atomicAdd(o + 3, v.w * nrm);
}

// ---------- BatchNorm(eval) + bias + ReLU
__global__ void k_bn_relu(const float* __restrict__ agg, const float* __restrict__ b,
                          const float* __restrict__ g, const float* __restrict__ be,
                          const float* __restrict__ m, const float* __restrict__ v,
                          float* __restrict__ x, int F, long long total) {
    long long i = blockIdx.x * (long long)blockDim.x + threadIdx.x;
    if (i >= total) return;
    int f = (int)(i % F);
    float val = (agg[i] + b[f] - m[f]) * rsqrtf(v[f] + EPS) * g[f] + be[f];
    x[i] = val > 0.0f ? val : 0.0f;
}

// ---------- GAT: per-node per-head attention logits
__global__ void k_gat_logits(const float* __restrict__ hg, const float* __restrict__ att_src,
                             const float* __restrict__ att_dst,
                             float* __restrict__ as_, float* __restrict__ ad_, int N) {
    int i = blockIdx.x * blockDim.x + threadIdx.x;
    if (i >= N * 4) return;
    int n = i >> 2, h = i & 3;
    const float* hp = hg + (long long)n * 64 + h * 16;
    const float* sp = att_src + h * 16;
    const float* dp = att_dst + h * 16;
    float accs = 0.0f, accd = 0.0f;
#pragma unroll
    for (int f = 0; f < 16; ++f) {
        float hv = hp[f];
        accs += hv * sp[f];
        accd += hv * dp[f];
    }
    as_[i] = accs;
    ad_[i] = accd;
}

__global__ void k_gat_init(unsigned* __restrict__ emax, float* __restrict__ denom, int NH) {
    int i = blockIdx.x * blockDim.x + threadIdx.x;
    if (i < NH) { emax[i] = 0u; denom[i] = 0.0f; }   // enc-space minimum
}

__device__ __forceinline__ float gat_logit(const float* as_, const float* ad_,
                                           int s, int d, int h) {
    float l = as_[s * 4 + h] + ad_[d * 4 + h];
    return l > 0.0f ? l : NEG_SLOPE * l;
}

__device__ __forceinline__ void edge_sd(int e, int E, const int* src, const int* dst,
                                        int& s, int& d) {
    if (e < E) { s = src[e]; d = dst[e]; } else { s = d = e - E; }
}

__global__ void k_gat_max(const int* __restrict__ src, const int* __restrict__ dst,
                          const float* __restrict__ as_, const float* __restrict__ ad_,
                          unsigned* __restrict__ emax, int E, int N) {
    long long t = blockIdx.x * (long long)blockDim.x + threadIdx.x;
    if (t >= (long long)(E + N) * 4) return;
    int e = (int)(t >> 2), h = (int)(t & 3);
    int s, d; edge_sd(e, E, src, dst, s, d);
    float l = gat_logit(as_, ad_, s, d, h);
    atomicMax(&emax[d * 4 + h], ordenc(l));
}

__global__ void k_gat_sum(const int* __restrict__ src, const int* __restrict__ dst,
                          const float* __restrict__ as_, const float* __restrict__ ad_,
                          const unsigned* __restrict__ emax, float* __restrict__ denom,
                          int E, int N) {
    long long t = blockIdx.x * (long long)blockDim.x + threadIdx.x;
    if (t >= (long long)(E + N) * 4) return;
    int e = (int)(t >> 2), h = (int)(t & 3);
    int s, d; edge_sd(e, E, src, dst, s, d);
    float l = gat_logit(as_, ad_, s, d, h);
    float ee = expf(l - orddec(emax[d * 4 + h]));
    atomicAdd(&denom[d * 4 + h], ee);
}

__global__ void k_gat_agg(const int* __restrict__ src, const int* __restrict__ dst,
                          const float* __restrict__ as_, const float* __restrict__ ad_,
                          const unsigned* __restrict__ emax, const float* __restrict__ denom,
                          const float* __restrict__ hg, float* __restrict__ out,
                          int E, int N) {
    long long t = blockIdx.x * (long long)blockDim.x + threadIdx.x;
    if (t >= (long long)(E + N) * 16) return;
    int e = (int)(t >> 4);
    int r = (int)(t & 15);
    int h = r >> 2, c = r & 3;
    int s, d; edge_sd(e, E, src, dst, s, d);
    int dh = d * 4 + h;
    float l = gat_logit(as_, ad_, s, d, h);
    float alpha = expf(l - orddec(emax[dh])) / denom[dh];
    float4 v = *(const float4*)(hg + (long long)s * 64 + h * 16 + c * 4);
    float* o = out + (long long)d * 64 + h * 16 + c * 4;
    atomicAdd(o + 0, v.x * alpha);
    atomicAdd(o + 1, v.y * alpha);
    atomicAdd(o + 2, v.z * alpha);
    atomicAdd(o + 3, v.w * alpha);
}

// ---------- head mean + bias + ReLU + log_softmax ----------
__global__ void k_gat_out(const float* __restrict__ acc, const float* __restrict__ bg,
                          float* __restrict__ out, int N) {
    int n = blockIdx.x * blockDim.x + threadIdx.x;
    if (n >= N) return;
    const float* a = acc + (long long)n * 64;
    float y[16];
    float mx = -1e30f;
#pragma unroll
    for (int f = 0; f < 16; ++f) {
        float v = 0.25f * (a[f] + a[16 + f] + a[32 + f] + a[48 + f]) + bg[f];
        v = v > 0.0f ? v : 0.0f;
        y[f] = v;
        mx = v > mx ? v : mx;
    }
    float sum = 0.0f;
#pragma unroll
    for (int f = 0; f < 16; ++f) sum += expf(y[f] - mx);
    float lse = mx + logf(sum);
    float* o = out + (long long)n * 16;
#pragma unroll
    for (int f = 0; f < 16; ++f) o[f] = y[f] - lse;
}

// ======================================================================
static inline unsigned nblk(long long n, int bs) { return (unsigned)((n + bs - 1) / bs); }

extern "C" void kernel_launch(void* const* d_in, const int* in_sizes, int n_in,
                              void* d_out, int out_size, void* d_ws, size_t ws_size,
                              hipStream_t stream) {
    const int N = in_sizes[0] / D_IN;
    const int E = in_sizes[1] / 2;

    const float* x   = (const float*)d_in[0];
    const int* eidx  = (const int*)d_in[1];
    const int* src   = eidx;
    const int* dst   = eidx + E;
    const float* W1  = (const float*)d_in[2];   const float* b1 = (const float*)d_in[3];
    const float* W2  = (const float*)d_in[4];   const float* b2 = (const float*)d_in[5];
    const float* W3  = (const float*)d_in[6];   const float* b3 = (const float*)d_in[7];
    const float* g1  = (const float*)d_in[8];   const float* be1 = (const float*)d_in[9];
    const float* m1  = (const float*)d_in[10];  const float* v1  = (const float*)d_in[11];
    const float* g2  = (const float*)d_in[12];  const float* be2 = (const float*)d_in[13];
    const float* m2  = (const float*)d_in[14];  const float* v2  = (const float*)d_in[15];
    const float* g3  = (const float*)d_in[16];  const float* be3 = (const float*)d_in[17];
    const float* m3  = (const float*)d_in[18];  const float* v3  = (const float*)d_in[19];
    const float* Wg  = (const float*)d_in[20];
    const float* att_src = (const float*)d_in[21];
    const float* att_dst = (const float*)d_in[22];
    const float* bg  = (const float*)d_in[23];

    // workspace carve-up (floats)
    float* ws     = (float*)d_ws;
    float* dis    = ws;                              // N   (u32 deg first, then f32 dis)
    float* bufH   = dis + N;                         // N*64 (GEMM outputs)
    float* bufAgg = bufH + (long long)N * 64;        // N*64 (scatter accumulators)
    float* bufX   = bufAgg + (long long)N * 64;      // N*64 (activations)
    float* as_    = bufX + (long long)N * 64;        // N*4
    float* ad_    = as_ + (long long)N * 4;          // N*4
    unsigned* emaxu = (unsigned*)(ad_ + (long long)N * 4); // N*4
    float* denom  = (float*)emaxu + (long long)N * 4;      // N*4

    const int BS = 256;
    const int Mtiles = N / 16;

    // --- degrees / symmetric normalization ---
    k_init_deg<<<nblk(N, BS), BS, 0, stream>>>((unsigned*)dis, N);
    k_count_deg<<<nblk(E, BS), BS, 0, stream>>>(dst, (unsigned*)dis, E);
    k_deg_to_dis<<<nblk(N, BS), BS, 0, stream>>>(dis, N);

    const long long EN = (long long)E + N;

    // --- GCN layer 1: 128 -> 64 ---
    {
        const int K = 128, F = 64;
        long long waves = (long long)Mtiles * (F / 16);
        k_wmma_gemm_f32<<<nblk(waves, 8), BS, 0, stream>>>(x, W1, bufH, Mtiles, K, F);
        k_zero<<<nblk((long long)N * F, BS), BS, 0, stream>>>(bufAgg, (long long)N * F);
        k_gcn_scatter<<<nblk(EN * (F / 4), BS), BS, 0, stream>>>(bufH, src, dst, dis, bufAgg, F / 4, E, N);
        k_bn_relu<<<nblk((long long)N * F, BS), BS, 0, stream>>>(bufAgg, b1, g1, be1, m1, v1, bufX, F, (long long)N * F);
    }
    // --- GCN layer 2: 64 -> 32 ---
    {
        const int K = 64, F = 32;
        long long waves = (long long)Mtiles * (F / 16);
        k_wmma_gemm_f32<<<nblk(waves, 8), BS, 0, stream>>>(bufX, W2, bufH, Mtiles, K, F);
        k_zero<<<nblk((long long)N * F, BS), BS, 0, stream>>>(bufAgg, (long long)N * F);
        k_gcn_scatter<<<nblk(EN * (F / 4), BS), BS, 0, stream>>>(bufH, src, dst, dis, bufAgg, F / 4, E, N);
        k_bn_relu<<<nblk((long long)N * F, BS), BS, 0, stream>>>(bufAgg, b2, g2, be2, m2, v2, bufX, F, (long long)N * F);
    }
    // --- GCN layer 3: 32 -> 16 ---
    {
        const int K = 32, F = 16;
        long long waves = (long long)Mtiles * (F / 16);
        k_wmma_gemm_f32<<<nblk(waves, 8), BS, 0, stream>>>(bufX, W3, bufH, Mtiles, K, F);
        k_zero<<<nblk((long long)N * F, BS), BS, 0, stream>>>(bufAgg, (long long)N * F);
        k_gcn_scatter<<<nblk(EN * (F / 4), BS), BS, 0, stream>>>(bufH, src, dst, dis, bufAgg, F / 4, E, N);
        k_bn_relu<<<nblk((long long)N * F, BS), BS, 0, stream>>>(bufAgg, b3, g3, be3, m3, v3, bufX, F, (long long)N * F);
    }

    // --- GAT: h = x3 @ Wg  (16 -> 64, viewed as [N,4,16]) ---
    {
        const int K = 16, F = 64;
        long long waves = (long long)Mtiles * (F / 16);
        k_wmma_gemm_f32<<<nblk(waves, 8), BS, 0, stream>>>(bufX, Wg, bufH, Mtiles, K, F);
    }
    k_gat_logits<<<nblk((long long)N * 4, BS), BS, 0, stream>>>(bufH, att_src, att_dst, as_, ad_, N);
    k_gat_init<<<nblk((long long)N * 4, BS), BS, 0, stream>>>(emaxu, denom, N * 4);
    k_zero<<<nblk((long long)N * 64, BS), BS, 0, stream>>>(bufAgg, (long long)N * 64);
    k_gat_max<<<nblk(EN * 4, BS), BS, 0, stream>>>(src, dst, as_, ad_, emaxu, E, N);
    k_gat_sum<<<nblk(EN * 4, BS), BS, 0, stream>>>(src, dst, as_, ad_, emaxu, denom, E, N);
    k_gat_agg<<<nblk(EN * 16, BS), BS, 0, stream>>>(src, dst, as_, ad_, emaxu, denom, bufH, bufAgg, E, N);
    k_gat_out<<<nblk(N, BS), BS, 0, stream>>>(bufAgg, bg, (float*)d_out, N);
}